// GATLayer_14886356648762
// MI455X (gfx1250) — compile-verified
//
#include <hip/hip_runtime.h>
#include <cstdint>
#include <cstddef>

#define N_NODES 50000
#define N_EDGES 1600000
#define HEADS 4
#define IN_FEAT 256
#define OUT_FEAT 32
#define NEG_SLOPE 0.2f
#define EPS_F 1e-8f

typedef __attribute__((ext_vector_type(2))) float v2f;
typedef __attribute__((ext_vector_type(8))) float v8f;

// Only meaningful to check on the device pass; the host (x86) pass does not
// register amdgcn builtins.
#if defined(__HIP_DEVICE_COMPILE__) && !__has_builtin(__builtin_amdgcn_wmma_f32_16x16x4_f32)
#error "gfx1250 wmma f32 16x16x4 builtin not available on this toolchain (device pass)"
#endif

// Monotone float <-> uint mapping so unsigned atomicMax implements float max.
__device__ __forceinline__ unsigned enc_ordered(float f) {
    unsigned b = __float_as_uint(f);
    return (b & 0x80000000u) ? ~b : (b | 0x80000000u);
}
__device__ __forceinline__ float dec_ordered(unsigned u) {
    unsigned b = (u & 0x80000000u) ? (u & 0x7FFFFFFFu) : ~u;
    return __uint_as_float(b);
}

// ---------------- kernel 0: init output / e_sum / global max ----------------
__global__ void gat_init(float* __restrict__ out, float* __restrict__ esum,
                         unsigned* __restrict__ gmax) {
    size_t i = (size_t)blockIdx.x * blockDim.x + threadIdx.x;
    if (i < (size_t)N_NODES * 128) out[i] = 0.0f;
    if (i < (size_t)N_NODES * HEADS) esum[i] = 0.0f;
    if (i == 0) *gmax = 0x007FFFFFu;  // enc_ordered(-inf)
}

// ---------------- kernel 1: h = x @ W via f32 WMMA ----------------
// Block = 256 threads = 8 waves. Block owns 16 rows of x (staged in LDS),
// wave w owns output columns [16w, 16w+16). K=256 swept in steps of 4 with
// v_wmma_f32_16x16x4_f32 accumulation.
__global__ __launch_bounds__(256) void gat_gemm(const float* __restrict__ x,
                                                const float* __restrict__ W,
                                                float* __restrict__ hbuf) {
    __shared__ float lds[16 * 260];  // stride 260 => 16 column-readers hit distinct banks
    const int t = threadIdx.x;
    const int row0 = blockIdx.x * 16;  // 50000 / 16 == 3125 exactly

    // cooperative load: 16 x 256 f32 tile = 1024 float4, 4 per thread, coalesced
#pragma unroll
    for (int j = 0; j < 4; ++j) {
        int f = t + 256 * j;
        int r = f >> 6;       // 64 float4 per row
        int c4 = f & 63;
        float4 v = ((const float4*)(x + (size_t)(row0 + r) * IN_FEAT))[c4];
        *(float4*)&lds[r * 260 + c4 * 4] = v;
    }
    __syncthreads();

    const int lane = t & 31;
    const int w = t >> 5;          // wave id 0..7 -> N-tile
    const int m = lane & 15;       // A: row M, B/C/D: column N
    const int khi = lane >> 4;     // A/B: K sub-block select (K = 2*khi + vgpr)
    const int col = w * 16 + m;
    const int head = col >> 5;
    const int o = col & 31;
    const float* Wp = W + (size_t)head * (IN_FEAT * OUT_FEAT) + o;

    v8f c = {0.f, 0.f, 0.f, 0.f, 0.f, 0.f, 0.f, 0.f};
#pragma unroll 4
    for (int k0 = 0; k0 < IN_FEAT; k0 += 4) {
        const int k = k0 + 2 * khi;
        v2f a;
        a.x = lds[m * 260 + k];
        a.y = lds[m * 260 + k + 1];
        v2f b;
        b.x = Wp[(size_t)k * OUT_FEAT];
        b.y = Wp[(size_t)(k + 1) * OUT_FEAT];
        c = __builtin_amdgcn_wmma_f32_16x16x4_f32(false, a, false, b, (short)0, c,
                                                  false, false);
    }

    // C layout: VGPR r holds row M = r + 8*khi at column N = lane&15
    float* hp = hbuf + (size_t)row0 * 128 + w * 16 + m;
    const int mb = khi * 8;
#pragma unroll
    for (int r = 0; r < 8; ++r) hp[(size_t)(mb + r) * 128] = c[r];
}

// ---------------- kernel 2: attn_l/attn_r = h . a ----------------
__global__ void gat_attn(const float* __restrict__ hbuf, const float* __restrict__ a_l,
                         const float* __restrict__ a_r, float* __restrict__ attn_l,
                         float* __restrict__ attn_r) {
    int i = blockIdx.x * blockDim.x + threadIdx.x;  // n*HEADS + h
    if (i >= N_NODES * HEADS) return;
    int n = i >> 2, hd = i & 3;
    const float4* hp = (const float4*)(hbuf + (size_t)n * 128 + hd * 32);
    const float4* lp = (const float4*)(a_l + hd * 32);
    const float4* rp = (const float4*)(a_r + hd * 32);
    float sl = 0.f, sr = 0.f;
#pragma unroll
    for (int j = 0; j < 8; ++j) {
        float4 h4 = hp[j], l4 = lp[j], r4 = rp[j];
        sl += h4.x * l4.x + h4.y * l4.y + h4.z * l4.z + h4.w * l4.w;
        sr += h4.x * r4.x + h4.y * r4.y + h4.z * r4.z + h4.w * r4.w;
    }
    attn_l[i] = sl;
    attn_r[i] = sr;
}

// ---------------- kernel 3: global max of leaky-relu(e) ----------------
__global__ __launch_bounds__(256) void gat_emax(const long long* __restrict__ ei,
                                                const float* __restrict__ attn_l,
                                                const float* __restrict__ attn_r,
                                                unsigned* __restrict__ gmax) {
    __shared__ float red[8];
    float mx = -3.402823466e38f;
    for (size_t i = (size_t)blockIdx.x * blockDim.x + threadIdx.x; i < N_EDGES;
         i += (size_t)gridDim.x * blockDim.x) {
        long long s = ei[i], d = ei[N_EDGES + i];
        float4 al = ((const float4*)attn_l)[s];
        float4 ar = ((const float4*)attn_r)[d];
        float e0 = al.x + ar.x, e1 = al.y + ar.y, e2 = al.z + ar.z, e3 = al.w + ar.w;
        e0 = e0 >= 0.f ? e0 : NEG_SLOPE * e0;
        e1 = e1 >= 0.f ? e1 : NEG_SLOPE * e1;
        e2 = e2 >= 0.f ? e2 : NEG_SLOPE * e2;
        e3 = e3 >= 0.f ? e3 : NEG_SLOPE * e3;
        mx = fmaxf(mx, fmaxf(fmaxf(e0, e1), fmaxf(e2, e3)));
    }
#pragma unroll
    for (int off = 16; off; off >>= 1) mx = fmaxf(mx, __shfl_xor(mx, off, 32));
    if ((threadIdx.x & 31) == 0) red[threadIdx.x >> 5] = mx;
    __syncthreads();
    if (threadIdx.x == 0) {
        float m2 = red[0];
#pragma unroll
        for (int j = 1; j < 8; ++j) m2 = fmaxf(m2, red[j]);
        atomicMax(gmax, enc_ordered(m2));
    }
}

// ---------------- kernel 4: e_exp per edge + segment sums ----------------
__global__ void gat_eexp(const long long* __restrict__ ei, const float* __restrict__ attn_l,
                         const float* __restrict__ attn_r, const unsigned* __restrict__ gmax,
                         float* __restrict__ eexp, float* __restrict__ esum) {
    size_t i = (size_t)blockIdx.x * blockDim.x + threadIdx.x;
    if (i >= N_EDGES) return;
    const float gm = dec_ordered(*gmax);
    long long s = ei[i], d = ei[N_EDGES + i];
    float4 al = ((const float4*)attn_l)[s];
    float4 ar = ((const float4*)attn_r)[d];
    float e[4] = {al.x + ar.x, al.y + ar.y, al.z + ar.z, al.w + ar.w};
    float4 ex;
    float* exv = (float*)&ex;
#pragma unroll
    for (int hd = 0; hd < 4; ++hd) {
        float v = e[hd];
        v = v >= 0.f ? v : NEG_SLOPE * v;
        float xv = expf(v - gm);
        exv[hd] = xv;
        atomicAdd(&esum[(size_t)d * 4 + hd], xv);
    }
    ((float4*)eexp)[i] = ex;
}

// ---------------- kernel 5: out[dst] += alpha * h[src] ----------------
// One thread per float4 of message payload: E * 32 threads.
__global__ void gat_agg(const long long* __restrict__ ei, const float* __restrict__ hbuf,
                        const float* __restrict__ eexp, const float* __restrict__ esum,
                        float* __restrict__ out) {
    size_t i = (size_t)blockIdx.x * blockDim.x + threadIdx.x;
    if (i >= (size_t)N_EDGES * 32) return;
    size_t e = i >> 5;
    int q = (int)(i & 31);
    int hd = q >> 3, o4 = q & 7;
    long long s = ei[e], d = ei[N_EDGES + e];
    float alpha = eexp[e * 4 + hd] / (esum[(size_t)d * 4 + hd] + EPS_F);
    float4 hv = *(const float4*)(hbuf + (size_t)s * 128 + hd * 32 + o4 * 4);
    float* op = out + (size_t)d * 128 + hd * 32 + o4 * 4;
    atomicAdd(op + 0, alpha * hv.x);
    atomicAdd(op + 1, alpha * hv.y);
    atomicAdd(op + 2, alpha * hv.z);
    atomicAdd(op + 3, alpha * hv.w);
}

extern "C" void kernel_launch(void* const* d_in, const int* in_sizes, int n_in,
                              void* d_out, int out_size, void* d_ws, size_t ws_size,
                              hipStream_t stream) {
    const float* x = (const float*)d_in[0];
    const long long* ei = (const long long*)d_in[1];  // int64 (2, E)
    const float* W = (const float*)d_in[2];
    const float* a_l = (const float*)d_in[3];
    const float* a_r = (const float*)d_in[4];
    float* out = (float*)d_out;

    // Workspace layout (all 16B aligned): h | e_exp | attn_l | attn_r | e_sum | gmax
    char* ws = (char*)d_ws;
    float* hbuf = (float*)ws;                                           // N*128
    float* eexp = (float*)(ws + (size_t)N_NODES * 128 * 4);             // E*4
    float* attn_l = (float*)(ws + (size_t)N_NODES * 128 * 4 + (size_t)N_EDGES * 4 * 4);
    float* attn_r = attn_l + (size_t)N_NODES * HEADS;
    float* esum = attn_r + (size_t)N_NODES * HEADS;
    unsigned* gmax = (unsigned*)(esum + (size_t)N_NODES * HEADS);

    gat_init<<<(N_NODES * 128 + 255) / 256, 256, 0, stream>>>(out, esum, gmax);
    gat_gemm<<<N_NODES / 16, 256, 0, stream>>>(x, W, hbuf);
    gat_attn<<<(N_NODES * HEADS + 255) / 256, 256, 0, stream>>>(hbuf, a_l, a_r, attn_l, attn_r);
    gat_emax<<<2048, 256, 0, stream>>>(ei, attn_l, attn_r, gmax);
    gat_eexp<<<(N_EDGES + 255) / 256, 256, 0, stream>>>(ei, attn_l, attn_r, gmax, eexp, esum);
    gat_agg<<<(int)(((size_t)N_EDGES * 32 + 255) / 256), 256, 0, stream>>>(ei, hbuf, eexp, esum, out);
}